// TorchGrouper_56719338111372
// MI455X (gfx1250) — compile-verified
//
#include <hip/hip_runtime.h>

// Problem constants (from reference)
#define NMAP 2
#define ZD   40
#define YD   400
#define XD   400
#define G_NUM 30000
#define M_NUM 120000
#define C_NUM 64
#define K_NUM 64
#define GK      (G_NUM * K_NUM)        // 1,920,000 (g,k) pairs
#define SF_SIZE (C_NUM * GK)           // 122,880,000 floats
#define GPF_SIZE (4 * GK)              // 7,680,000 floats
#define PAIRS_PER_BLOCK 256
#define ROW_STRIDE 68                  // floats; 272B row: 16B aligned, skewed banks

__global__ __launch_bounds__(PAIRS_PER_BLOCK)
void grouper_kernel(const int* __restrict__ voxel_maps,
                    const int* __restrict__ grid_positions,
                    const float* __restrict__ features,
                    const int* __restrict__ index_offset,
                    float* __restrict__ out)
{
    __shared__ float srow[PAIRS_PER_BLOCK * ROW_STRIDE];  // 69,632 B staged rows
    __shared__ int   nonempty[4];

    const int tid = threadIdx.x;
    const int p   = blockIdx.x * PAIRS_PER_BLOCK + tid;   // flat (g,k)
    const int g   = p >> 6;
    const int k   = p & 63;
    const int gl  = tid >> 6;                             // local g (0..3)

    if (tid < 4) nonempty[tid] = 0;

    // positions: grid row [n,z,y,x] + offset row [0,rx,ry,rz]
    const int4 gp  = ((const int4*)grid_positions)[g];
    const int4 off = ((const int4*)index_offset)[k];

    const int n = gp.x;                                    // offset comp0 == 0
    int z = gp.y + off.y; z = min(max(z, 0), ZD - 1);
    int y = gp.z + off.z; y = min(max(y, 0), YD - 1);
    int x = gp.w + off.w; x = min(max(x, 0), XD - 1);

    const int vi  = ((n * ZD + z) * YD + y) * XD + x;
    const int idx = voxel_maps[vi];

    // --- stage this pair's 256B feature row into LDS via CDNA5 async DMA ---
    float* rowp = &srow[tid * ROW_STRIDE];
    if (idx >= 0) {
        const float* src = features + (size_t)idx * C_NUM;
        // generic LDS pointer: low 32 bits are the LDS byte offset (shared
        // aperture occupies ADDR[63:32] only, per CDNA5 aperture rules)
        unsigned lds_base = (unsigned)(size_t)rowp;
        unsigned long long gaddr = (unsigned long long)(size_t)src;
        #pragma unroll
        for (int q = 0; q < 16; ++q) {                     // 16 x 16B = 256B row
            asm volatile("global_load_async_to_lds_b128 %0, %1, off"
                         :
                         : "v"(lds_base + (unsigned)(q * 16)),
                           "v"(gaddr + (unsigned long long)(q * 16))
                         : "memory");
        }
    } else {
        #pragma unroll
        for (int q = 0; q < 16; ++q)
            ((float4*)rowp)[q] = make_float4(0.f, 0.f, 0.f, 0.f);
    }

    // --- gpf while async copies are in flight: gpf[0,comp,g,k] = off[comp] ---
    // NT stores: write-once stream, keep L2 for the gather tables.
    float* gout = out + SF_SIZE;
    __builtin_nontemporal_store((float)off.x, &gout[(size_t)0 * GK + p]);  // 0
    __builtin_nontemporal_store((float)off.y, &gout[(size_t)1 * GK + p]);
    __builtin_nontemporal_store((float)off.z, &gout[(size_t)2 * GK + p]);
    __builtin_nontemporal_store((float)off.w, &gout[(size_t)3 * GK + p]);

    __syncthreads();                               // nonempty[] zeroed
    if (idx >= 0) atomicOr(&nonempty[gl], 1);

    asm volatile("s_wait_asynccnt 0" ::: "memory"); // own rows landed in LDS
    __syncthreads();                                // atomics visible

    // --- transposed write-out: out[c*GK + p] = row[c]; fully coalesced NT ---
    #pragma unroll
    for (int c4 = 0; c4 < 16; ++c4) {
        float4 v = ((const float4*)rowp)[c4];
        __builtin_nontemporal_store(v.x, &out[(size_t)(4 * c4 + 0) * GK + p]);
        __builtin_nontemporal_store(v.y, &out[(size_t)(4 * c4 + 1) * GK + p]);
        __builtin_nontemporal_store(v.z, &out[(size_t)(4 * c4 + 2) * GK + p]);
        __builtin_nontemporal_store(v.w, &out[(size_t)(4 * c4 + 3) * GK + p]);
    }

    // --- empty mask: True (1.0f) when all 64 k's were empty ---
    if (tid < 4) {
        const int g0 = blockIdx.x * 4 + tid;
        __builtin_nontemporal_store(nonempty[tid] ? 0.0f : 1.0f,
                                    &out[(size_t)SF_SIZE + (size_t)GPF_SIZE + g0]);
    }
}

extern "C" void kernel_launch(void* const* d_in, const int* in_sizes, int n_in,
                              void* d_out, int out_size, void* d_ws, size_t ws_size,
                              hipStream_t stream) {
    const int*   voxel_maps     = (const int*)d_in[0];
    const int*   grid_positions = (const int*)d_in[1];
    const float* features       = (const float*)d_in[2];
    const int*   index_offset   = (const int*)d_in[3];
    float*       out            = (float*)d_out;

    (void)in_sizes; (void)n_in; (void)out_size; (void)d_ws; (void)ws_size;

    dim3 grid(GK / PAIRS_PER_BLOCK);   // 7500 blocks, exact cover
    grouper_kernel<<<grid, PAIRS_PER_BLOCK, 0, stream>>>(
        voxel_maps, grid_positions, features, index_offset, out);
}